// GUMSampler_20615843021196
// MI455X (gfx1250) — compile-verified
//
#include <hip/hip_runtime.h>

typedef __attribute__((ext_vector_type(2))) float v2f;
typedef __attribute__((ext_vector_type(8))) float v8f;

#define KC 16   // number of clusters (K)
#define KD 15   // embedding dim (K-1)
#define TILES_PER_WAVE 16
#define WAVES_PER_BLOCK 8
#define BLOCK_THREADS 256

// ds_swizzle_b32 group-of-32 encoding: offset[14:10]=xor, [9:5]=or, [4:0]=and
// SWAPX16: xor=0x10, or=0x00, and=0x1F  ->  0x401F
#define SWZ_SWAPX16 0x401F

static __device__ __forceinline__ float swz16f(float x) {
    return __uint_as_float(
        (unsigned)__builtin_amdgcn_ds_swizzle(__float_as_int(x), SWZ_SWAPX16));
}
static __device__ __forceinline__ unsigned swz16u(unsigned x) {
    return (unsigned)__builtin_amdgcn_ds_swizzle((int)x, SWZ_SWAPX16);
}
static __device__ __forceinline__ unsigned uminu(unsigned a, unsigned b) {
    return a < b ? a : b;
}

// cross = V(16x15, zero-padded to 16x16) @ Z(16xN) via V_WMMA_F32_16X16X4_F32
// chained 4x (K=16). Argmin is done on d^2 packed with the cluster index into
// a uint (d^2 >= 0 so float bits order as unsigned; low 4 bits carry the index
// => exact lowest-index tie-break). One hardware sqrt per pixel at the end.
__global__ __launch_bounds__(BLOCK_THREADS)
void gum_sampler_wmma(const float* __restrict__ z,      // (KD, N) row-major
                      const float* __restrict__ verts,  // (KC, KD) row-major
                      int*   __restrict__ Xout,         // (N) int32 argmin
                      float* __restrict__ dmin,         // (N) f32 min distance
                      long long N)
{
    const int lane = threadIdx.x & 31;
    const int wave = threadIdx.x >> 5;
    const int hi   = lane >> 4;     // 0 = lanes 0-15, 1 = lanes 16-31
    const int lm   = lane & 15;

    // ---- A matrix (wave-invariant): 16x4 f32 per WMMA step t ----
    // ISA layout: lanes 0-15 hold M=lane, {K0,K1}; lanes 16-31 hold M=lane-16, {K2,K3}
    v2f a[4];
#pragma unroll
    for (int t = 0; t < 4; ++t) {
        int k0 = 4 * t + 2 * hi;
        int k1 = k0 + 1;
        a[t].x = (k0 < KD) ? verts[lm * KD + k0] : 0.0f;   // K=15 pad column -> 0
        a[t].y = (k1 < KD) ? verts[lm * KD + k1] : 0.0f;
    }

    // ---- ||v_k||^2 for this lane-half's 8 clusters (k = r + 8*hi) ----
    float vv[8];
#pragma unroll
    for (int r = 0; r < 8; ++r) {
        int k = r + 8 * hi;
        float s = 0.0f;
#pragma unroll
        for (int j = 0; j < KD; ++j) {
            float v = verts[k * KD + j];
            s = fmaf(v, v, s);
        }
        vv[r] = s;
    }

    const long long gw   = (long long)blockIdx.x * WAVES_PER_BLOCK + wave;
    const long long base = gw * (16LL * TILES_PER_WAVE);
    if (base >= N) return;                       // wave-uniform

    // ---- per-lane row base pointers (hoisted): 8 z-rows this lane loads ----
    // Row K=15 doesn't exist: clamp to row 14 (A's K=15 column is zero, so the
    // WMMA contribution vanishes; the value is zeroed below for ||z||^2).
    const float* rp[8];
#pragma unroll
    for (int j = 0; j < 8; ++j) {
        int k  = 4 * (j >> 1) + 2 * hi + (j & 1);
        int kc = (k < KD) ? k : (KD - 1);
        rp[j] = z + (long long)kc * N + base + lm;
    }
    int*   Xp = Xout + base + lm;
    float* Dp = dmin + base + lm;
    const unsigned idx0 = 8u * (unsigned)hi;

    for (int it = 0; it < TILES_PER_WAVE; ++it) {
        if (base + (long long)it * 16 >= N) return;   // wave-uniform guard
        const int off = it * 16;

        // ---- branchless B-tile load (one clause, immediate offsets) ----
        float bl[8];
#pragma unroll
        for (int j = 0; j < 8; ++j) bl[j] = rp[j][off];
        bl[7] = hi ? 0.0f : bl[7];                    // zero the K=15 pad row

        float zzp = 0.0f;
#pragma unroll
        for (int j = 0; j < 8; ++j) zzp = fmaf(bl[j], bl[j], zzp);

        v2f b[4];
#pragma unroll
        for (int t = 0; t < 4; ++t) { b[t].x = bl[2 * t]; b[t].y = bl[2 * t + 1]; }

        // ---- 16x16 cross tile: 4 chained fp32 WMMAs (K = 16) ----
        v8f c = {};
#pragma unroll
        for (int t = 0; t < 4; ++t) {
            c = __builtin_amdgcn_wmma_f32_16x16x4_f32(
                    /*neg_a=*/false, a[t], /*neg_b=*/false, b[t],
                    /*c_mod=*/(short)0, c, /*reuse_a=*/false, /*reuse_b=*/false);
        }

        // ---- ||z||^2: combine the two lane-halves (1 ds_swizzle) ----
        float zz = zzp + swz16f(zzp);

        // ---- packed argmin over this lane's 8 clusters ----
        unsigned best = 0xFFFFFFFFu;
#pragma unroll
        for (int r = 0; r < 8; ++r) {
            float d2 = fmaf(-2.0f, c[r], zz + vv[r]);
            d2 = fmaxf(d2, 0.0f);
            unsigned u = (__float_as_uint(d2) & 0xFFFFFFF0u) | (idx0 + (unsigned)r);
            best = uminu(best, u);
        }

        // ---- combine halves (1 ds_swizzle + 1 umin); ties -> lower index ----
        best = uminu(best, swz16u(best));

        const int   widx = (int)(best & 15u);
        const float dm   = __builtin_amdgcn_sqrtf(__uint_as_float(best & 0xFFFFFFF0u));

        if (hi == 0) {
            Xp[off] = widx;
            Dp[off] = dm;
        }
    }
}

extern "C" void kernel_launch(void* const* d_in, const int* in_sizes, int n_in,
                              void* d_out, int out_size, void* d_ws, size_t ws_size,
                              hipStream_t stream) {
    const float* z     = (const float*)d_in[0];   // (15, 2048, 2048) f32
    const float* verts = (const float*)d_in[1];   // (16, 15) f32

    const long long N = (long long)in_sizes[0] / KD;   // 2048*2048

    int*   Xout = (int*)d_out;            // first N elems: int32 argmin
    float* dmin = (float*)d_out + N;      // next  N elems: f32 min distance

    const long long pixelsPerBlock = 16LL * TILES_PER_WAVE * WAVES_PER_BLOCK; // 2048
    const int blocks = (int)((N + pixelsPerBlock - 1) / pixelsPerBlock);

    gum_sampler_wmma<<<blocks, BLOCK_THREADS, 0, stream>>>(z, verts, Xout, dmin, N);
}